// Set2Set_81587198755492
// MI455X (gfx1250) — compile-verified
//
#include <hip/hip_runtime.h>
#include <math.h>

#define DDIM   128
#define TWOD   256
#define FOURD  512
#define EPSV   1e-10f

typedef float v2f __attribute__((ext_vector_type(2)));
typedef float v8f __attribute__((ext_vector_type(8)));

// ---------------------------------------------------------------------------
// Zero-init q_star (d_out), h, c
// ---------------------------------------------------------------------------
__global__ void s2s_init(float* __restrict__ q_star, float* __restrict__ h,
                         float* __restrict__ c, int B) {
    int i = blockIdx.x * blockDim.x + threadIdx.x;
    if (i < B * TWOD) q_star[i] = 0.0f;
    if (i < B * DDIM) { h[i] = 0.0f; c[i] = 0.0f; }
}

// ---------------------------------------------------------------------------
// gates[B,512] = q_star[B,256] @ w_ih^T + h[B,128] @ w_hh^T + b_ih + b_hh
// One wave32 per 16x16 output tile, K stepped by 4 via V_WMMA_F32_16X16X4_F32.
//
// f32 16x16x4 fragment layout (ISA 7.12.2):
//   A (16x4): lane 0-15 -> M=lane, VGPR0/1 = K=0/1; lane 16-31 -> K=2/3
//   B (4x16): lane 0-15 -> N=lane, VGPR0/1 = K=0/1; lane 16-31 -> K=2/3
//   C/D     : VGPR r -> M = r (lanes 0-15) or r+8 (lanes 16-31), N = lane&15
// ---------------------------------------------------------------------------
__global__ __launch_bounds__(128)
void lstm_gates_wmma(const float* __restrict__ qstar,   // [B, 256]
                     const float* __restrict__ h,       // [B, 128]
                     const float* __restrict__ w_ih,    // [512, 256] row-major
                     const float* __restrict__ w_hh,    // [512, 128] row-major
                     const float* __restrict__ b_ih,    // [512]
                     const float* __restrict__ b_hh,    // [512]
                     float* __restrict__ gates,         // [B, 512]
                     int B)
{
    const int lane  = threadIdx.x & 31;
    const int wv    = threadIdx.x >> 5;
    const int tile  = blockIdx.x * 4 + wv;          // wave-uniform
    const int ntn   = FOURD / 16;                   // 32 tiles along N
    const int mt    = tile / ntn;
    const int nt    = tile % ntn;
    const int m0    = mt * 16;
    if (m0 >= B) return;                            // whole-wave uniform exit

    const int idx15 = lane & 15;                    // M row (for A) / N col (for B)
    const int kpair = (lane >> 4) * 2;              // 0 or 2
    const int n0    = nt * 16;

    const int arow  = (m0 + idx15 < B) ? (m0 + idx15) : (B - 1);  // clamp, guarded on store
    const int bcol  = n0 + idx15;

    // C initialized with broadcast bias (same N column in every C row register)
    const float bias = b_ih[bcol] + b_hh[bcol];
    v8f acc;
    #pragma unroll
    for (int r = 0; r < 8; ++r) acc[r] = bias;

    // ---- K over q_star (256) against w_ih ----
    const float* Ap = qstar + (size_t)arow * TWOD;
    const float* Bp = w_ih  + (size_t)bcol * TWOD;  // row j of w_ih == column j of w_ih^T
    for (int k0 = 0; k0 < TWOD; k0 += 4) {
        v2f a = *(const v2f*)(Ap + k0 + kpair);
        v2f b = *(const v2f*)(Bp + k0 + kpair);
        acc = __builtin_amdgcn_wmma_f32_16x16x4_f32(false, a, false, b,
                                                    (short)0, acc, false, false);
    }
    // ---- K over h (128) against w_hh ----
    const float* Ap2 = h    + (size_t)arow * DDIM;
    const float* Bp2 = w_hh + (size_t)bcol * DDIM;
    for (int k0 = 0; k0 < DDIM; k0 += 4) {
        v2f a = *(const v2f*)(Ap2 + k0 + kpair);
        v2f b = *(const v2f*)(Bp2 + k0 + kpair);
        acc = __builtin_amdgcn_wmma_f32_16x16x4_f32(false, a, false, b,
                                                    (short)0, acc, false, false);
    }

    // ---- store D ----
    const int mbase = m0 + ((lane >> 4) ? 8 : 0);
    #pragma unroll
    for (int r = 0; r < 8; ++r) {
        const int mrow = mbase + r;
        if (mrow < B) gates[(size_t)mrow * FOURD + n0 + idx15] = acc[r];
    }
}

// ---------------------------------------------------------------------------
// Pointwise LSTM update (PyTorch gate order i,f,g,o)
// ---------------------------------------------------------------------------
__global__ void lstm_pointwise(const float* __restrict__ gates,
                               float* __restrict__ h, float* __restrict__ c, int B)
{
    int idx = blockIdx.x * blockDim.x + threadIdx.x;
    if (idx >= B * DDIM) return;
    int b = idx / DDIM, d = idx - b * DDIM;
    const float* g = gates + (size_t)b * FOURD;
    float gi = g[d], gf = g[DDIM + d], gg = g[2 * DDIM + d], go = g[3 * DDIM + d];
    float i = 1.0f / (1.0f + __expf(-gi));
    float f = 1.0f / (1.0f + __expf(-gf));
    float t = tanhf(gg);
    float o = 1.0f / (1.0f + __expf(-go));
    float cn = f * c[idx] + i * t;
    c[idx] = cn;
    h[idx] = o * tanhf(cn);
}

// ---------------------------------------------------------------------------
// Per-graph segment attention + readout. One 8-wave block per graph.
// Wave-per-node, lane holds float4 slice of the 128-wide row.
// out = sum(e_n * x_n) / (sum(e_n) + eps)   (softmax normalization folded in)
// ---------------------------------------------------------------------------
__global__ __launch_bounds__(256)
void set2set_attention(const float* __restrict__ x,          // [N, 128]
                       const int*   __restrict__ node2graph, // [N] sorted
                       const float* __restrict__ h,          // [B, 128]
                       float*       __restrict__ prod,       // [N] scratch
                       float*       __restrict__ q_star,     // [B, 256] out
                       int N, int B)
{
    const int b    = blockIdx.x;
    const int tid  = threadIdx.x;
    const int lane = tid & 31;
    const int wv   = tid >> 5;

    __shared__ int   s_lo, s_hi;
    __shared__ float s_max[8];
    __shared__ float s_sum[8];
    __shared__ float s_acc[8 * DDIM];

    if (tid == 0) {
        int lo = 0, hi = N;                              // lower_bound(b)
        while (lo < hi) { int m = (lo + hi) >> 1; if (node2graph[m] < b) lo = m + 1; else hi = m; }
        s_lo = lo;
        int lo2 = lo, hi2 = N;                           // lower_bound(b+1)
        while (lo2 < hi2) { int m = (lo2 + hi2) >> 1; if (node2graph[m] < b + 1) lo2 = m + 1; else hi2 = m; }
        s_hi = lo2;
    }
    __syncthreads();
    const int lo = s_lo, hi = s_hi;

    const float4 hv = *(const float4*)(h + (size_t)b * DDIM + lane * 4);

    // ---- pass 1: logits + segment max ----
    float wmax = -INFINITY;
    for (int n = lo + wv; n < hi; n += 8) {
        const float4 xv = *(const float4*)(x + (size_t)n * DDIM + lane * 4);
        int pn = (n + 8 < hi) ? (n + 8) : n;
        __builtin_prefetch(x + (size_t)pn * DDIM + lane * 4, 0, 3);  // global_prefetch_b8
        float dot = hv.x * xv.x + hv.y * xv.y + hv.z * xv.z + hv.w * xv.w;
        #pragma unroll
        for (int off = 16; off >= 1; off >>= 1) dot += __shfl_xor(dot, off, 32);
        if (lane == 0) prod[n] = dot;
        wmax = fmaxf(wmax, dot);
    }
    if (lane == 0) s_max[wv] = wmax;
    __syncthreads();
    float m = -INFINITY;
    #pragma unroll
    for (int w = 0; w < 8; ++w) m = fmaxf(m, s_max[w]);

    // ---- pass 2: exp + weighted accumulation (normalize at the end) ----
    float4 acc = {0.f, 0.f, 0.f, 0.f};
    float sumE = 0.f;
    for (int n = lo + wv; n < hi; n += 8) {
        float e = __expf(prod[n] - m);
        const float4 xv = *(const float4*)(x + (size_t)n * DDIM + lane * 4);
        int pn = (n + 8 < hi) ? (n + 8) : n;
        __builtin_prefetch(x + (size_t)pn * DDIM + lane * 4, 0, 3);
        acc.x += e * xv.x; acc.y += e * xv.y; acc.z += e * xv.z; acc.w += e * xv.w;
        sumE += e;
    }
    s_acc[wv * DDIM + lane * 4 + 0] = acc.x;
    s_acc[wv * DDIM + lane * 4 + 1] = acc.y;
    s_acc[wv * DDIM + lane * 4 + 2] = acc.z;
    s_acc[wv * DDIM + lane * 4 + 3] = acc.w;
    if (lane == 0) s_sum[wv] = sumE;
    __syncthreads();

    if (tid < DDIM) {
        float v = 0.f, norm = 0.f;
        #pragma unroll
        for (int w = 0; w < 8; ++w) { v += s_acc[w * DDIM + tid]; norm += s_sum[w]; }
        q_star[(size_t)b * TWOD + DDIM + tid] = v / (norm + EPSV);   // readout half
        q_star[(size_t)b * TWOD + tid]        = h[(size_t)b * DDIM + tid]; // query half
    }
}

// ---------------------------------------------------------------------------
extern "C" void kernel_launch(void* const* d_in, const int* in_sizes, int n_in,
                              void* d_out, int out_size, void* d_ws, size_t ws_size,
                              hipStream_t stream)
{
    const float* x    = (const float*)d_in[0];
    const float* w_ih = (const float*)d_in[1];
    const float* w_hh = (const float*)d_in[2];
    const float* b_ih = (const float*)d_in[3];
    const float* b_hh = (const float*)d_in[4];
    const int*   n2g  = (const int*)d_in[5];
    // d_in[6] = batch_size scalar (on device; derive B on host instead)

    const int B = out_size / TWOD;          // q_star is [B, 2D]
    const int N = in_sizes[0] / DDIM;       // x is [N, D]

    float* gates = (float*)d_ws;                    // B*512
    float* h     = gates + (size_t)B * FOURD;       // B*128
    float* c     = h     + (size_t)B * DDIM;        // B*128
    float* prod  = c     + (size_t)B * DDIM;        // N
    float* qstar = (float*)d_out;                   // q_star lives in d_out across steps

    const int initN = B * TWOD;
    s2s_init<<<(initN + 255) / 256, 256, 0, stream>>>(qstar, h, c, B);

    const int mtiles = (B + 15) / 16;
    const int tiles  = mtiles * (FOURD / 16);
    const int gemm_blocks = (tiles + 3) / 4;        // 4 waves (tiles) per block

    for (int step = 0; step < 3; ++step) {
        lstm_gates_wmma<<<gemm_blocks, 128, 0, stream>>>(qstar, h, w_ih, w_hh,
                                                         b_ih, b_hh, gates, B);
        lstm_pointwise<<<(B * DDIM + 255) / 256, 256, 0, stream>>>(gates, h, c, B);
        set2set_attention<<<B, 256, 0, stream>>>(x, n2g, h, prod, qstar, N, B);
    }
}